// warp_feature_83296595738911
// MI455X (gfx1250) — compile-verified
//
#include <hip/hip_runtime.h>
#include <hip/hip_bf16.h>
#include <cstdint>
#include <cstddef>

// Problem constants (from the reference)
#define BB   8
#define CC   64
#define HH   256
#define WW   512
#define BOUND 64     // BOUNDARY, direction 'lr'

// One block = one row-tile of 256 consecutive x in one (b, y) row.
// Each thread owns one pixel (b,y,x): computes the 4 bilinear taps once and
// reuses them across the 64-channel loop. Warped + mask written with
// non-temporal stores (write-once 512MB stream; keep L2 for the target rows).
__global__ __launch_bounds__(256) void warp_feature_kernel(
    const float* __restrict__ disp,   // (B,1,H,W)
    const float* __restrict__ tgt,    // (B,C,H,W)
    float* __restrict__ warped,       // (B,C,H,W)
    float* __restrict__ mask)         // (B,C,H,W)
{
    __shared__ float sdisp[256];

    const int tid  = threadIdx.x;
    const int tile = blockIdx.x;          // 0 .. B*H*2 - 1
    const int xb   = (tile & 1) << 8;     // 0 or 256
    const int row  = tile >> 1;           // b*H + y
    const int y    = row & (HH - 1);
    const int b    = row >> 8;            // H == 256
    const int x    = xb + tid;

    // ---- CDNA5 async global->LDS staging of this wave's disp values ------
    // LDS byte address comes from one VGPR; global address is SGPR base plus VGPR byte offset.
    // Flat pointer to __shared__ has the LDS aperture in the high 32 bits,
    // so the low 32 bits are exactly the wave-relative LDS byte offset.
    {
        unsigned lds_off = (unsigned)(uintptr_t)(&sdisp[tid]);
        unsigned g_off   = (unsigned)((row * WW + x) * 4);
        asm volatile("global_load_async_to_lds_b32 %0, %1, %2"
                     :
                     : "v"(lds_off), "v"(g_off), "s"(disp)
                     : "memory");
        asm volatile("s_wait_asynccnt 0" ::: "memory");
    }
    const float d = sdisp[tid];           // each wave reads only its own lanes

    // ---- grid_sample(align_corners=False, padding 'zeros') math ----------
    const float shiftx = (float)(WW - 1) * 0.5f;
    const float shifty = (float)(HH - 1) * 0.5f;
    const float Xs  = (float)x + d;                        // direction 'lr'
    const float gx  = (Xs - shiftx) / shiftx;
    const float gy  = ((float)y - shifty) / shifty;
    const float ix  = ((gx + 1.0f) * (float)WW - 1.0f) * 0.5f;
    const float iy  = ((gy + 1.0f) * (float)HH - 1.0f) * 0.5f;

    const float x0f = floorf(ix);
    const float y0f = floorf(iy);
    const float fx  = ix - x0f;
    const float fy  = iy - y0f;
    const int x0 = (int)x0f, y0 = (int)y0f;
    const int x1 = x0 + 1,   y1 = y0 + 1;

    const float vx0 = (x0 >= 0 && x0 <= WW - 1) ? 1.0f : 0.0f;
    const float vx1 = (x1 >= 0 && x1 <= WW - 1) ? 1.0f : 0.0f;
    const float vy0 = (y0 >= 0 && y0 <= HH - 1) ? 1.0f : 0.0f;
    const float vy1 = (y1 >= 0 && y1 <= HH - 1) ? 1.0f : 0.0f;

    const int xc0 = min(max(x0, 0), WW - 1);
    const int xc1 = min(max(x1, 0), WW - 1);
    const int yc0 = min(max(y0, 0), HH - 1);
    const int yc1 = min(max(y1, 0), HH - 1);

    const float w00 = (1.0f - fx) * (1.0f - fy) * (vx0 * vy0);
    const float w01 = fx          * (1.0f - fy) * (vx1 * vy0);
    const float w10 = (1.0f - fx) * fy          * (vx0 * vy1);
    const float w11 = fx          * fy          * (vx1 * vy1);

    const size_t HWs  = (size_t)HH * WW;
    const float* bbas = tgt + (size_t)b * CC * HWs;
    const int i00 = yc0 * WW + xc0;
    const int i01 = yc0 * WW + xc1;
    const int i10 = yc1 * WW + xc0;
    const int i11 = yc1 * WW + xc1;

    // Warm the two sampled rows (gfx1250 global_prefetch_b8).
    __builtin_prefetch(bbas + i00, 0, 0);
    __builtin_prefetch(bbas + i10, 0, 0);

    const float mval = (x < (WW - BOUND)) ? 1.0f : 0.0f;
    size_t oidx = (((size_t)b * CC) * HH + (size_t)y) * WW + (size_t)x;  // c = 0

    #pragma unroll 4
    for (int c = 0; c < CC; ++c) {
        const float* pc = bbas + (size_t)c * HWs;
        const float v = w00 * pc[i00] + w01 * pc[i01]
                      + w10 * pc[i10] + w11 * pc[i11];
        __builtin_nontemporal_store(v,    warped + oidx);
        __builtin_nontemporal_store(mval, mask   + oidx);
        oidx += HWs;
    }
}

extern "C" void kernel_launch(void* const* d_in, const int* in_sizes, int n_in,
                              void* d_out, int out_size, void* d_ws, size_t ws_size,
                              hipStream_t stream) {
    const float* disp = (const float*)d_in[0];                // (B,1,H,W) fp32
    const float* tgt  = (const float*)d_in[1];                // (B,C,H,W) fp32
    float* warped = (float*)d_out;                            // output 0
    float* mask   = warped + (size_t)BB * CC * HH * WW;       // output 1 (0/1 floats)

    const int nblocks = BB * HH * (WW / 256);                 // 4096 blocks, 8 waves each
    warp_feature_kernel<<<nblocks, 256, 0, stream>>>(disp, tgt, warped, mask);
}